// MCPBRNN_Generic_PETconstraint_constantoutput_variableLoss_10196252360836
// MI455X (gfx1250) — compile-verified
//
#include <hip/hip_runtime.h>
#include <math.h>

typedef __attribute__((ext_vector_type(2))) float v2f;
typedef __attribute__((ext_vector_type(8))) float v8f;

#define C_ML 2.9086f
#define C_SL 1.898f

// ---------------------------------------------------------------------------
// Stage 1: per-block partial sum & sum-of-squares of y_obs[spin:train] using
// V_WMMA_F32_16X16X4_F32 with an all-ones B matrix (64 f32 / wave / wmma).
// Main loop is branch-free (full chunks, unconditional loads, EXEC always
// full for WMMA); the ragged tail is handled by wave 0 with address-clamped
// loads + value masking (v_cndmask, no exec divergence).
// Deterministic: each block writes its partial to a fixed ws slot.
// ---------------------------------------------------------------------------
__global__ void stats_partial_kernel(const float* __restrict__ y,
                                     const int* __restrict__ spinp,
                                     const int* __restrict__ trainp,
                                     float* __restrict__ ws) {
    const int spin  = spinp[0];
    const int train = trainp[0];
    const int n     = train - spin;

    __shared__ float s_sum[8];
    __shared__ float s_sq[8];

    const int lane = threadIdx.x & 31;
    const int wid  = threadIdx.x >> 5;
    const int wavesPerBlock = blockDim.x >> 5;
    const int globalWave    = blockIdx.x * wavesPerBlock + wid;
    const int nWaves        = gridDim.x * wavesPerBlock;

    v8f accS = {0.f,0.f,0.f,0.f,0.f,0.f,0.f,0.f};
    v8f accQ = {0.f,0.f,0.f,0.f,0.f,0.f,0.f,0.f};
    v2f ones; ones.x = 1.0f; ones.y = 1.0f;

    if (n > 0) {
        const int nFull = n >> 6;                    // full 64-element chunks
        // ---- branch-free main loop ----
        for (int chunk = globalWave; chunk < nFull; chunk += nWaves) {
            const int base = spin + (chunk << 6) + (lane << 1);
            const float x0 = y[base + 0];            // unconditional loads
            const float x1 = y[base + 1];
            v2f a;  a.x  = x0;      a.y  = x1;       // A fragment (16x4 f32)
            v2f a2; a2.x = x0 * x0; a2.y = x1 * x1;
            // D[m][n] += sum_k A[m][k]  (B == ones)
            accS = __builtin_amdgcn_wmma_f32_16x16x4_f32(
                       false, a,  false, ones, (short)0, accS, false, false);
            accQ = __builtin_amdgcn_wmma_f32_16x16x4_f32(
                       false, a2, false, ones, (short)0, accQ, false, false);
        }
        // ---- ragged tail: one wave, clamped addresses, masked values ----
        if ((n & 63) && globalWave == 0) {           // wave-uniform condition
            const int base = spin + (nFull << 6) + (lane << 1);
            const int i0 = (base + 0 < train) ? (base + 0) : (train - 1);
            const int i1 = (base + 1 < train) ? (base + 1) : (train - 1);
            float x0 = y[i0];                         // unconditional loads
            float x1 = y[i1];
            x0 = (base + 0 < train) ? x0 : 0.0f;      // v_cndmask masking
            x1 = (base + 1 < train) ? x1 : 0.0f;
            v2f a;  a.x  = x0;      a.y  = x1;
            v2f a2; a2.x = x0 * x0; a2.y = x1 * x1;
            accS = __builtin_amdgcn_wmma_f32_16x16x4_f32(
                       false, a,  false, ones, (short)0, accS, false, false);
            accQ = __builtin_amdgcn_wmma_f32_16x16x4_f32(
                       false, a2, false, ones, (short)0, accQ, false, false);
        }
    }

    // Every row-sum is replicated across the 16 columns of D -> divide by 16.
    float ls = (accS[0] + accS[1] + accS[2] + accS[3] +
                accS[4] + accS[5] + accS[6] + accS[7]) * 0.0625f;
    float lq = (accQ[0] + accQ[1] + accQ[2] + accQ[3] +
                accQ[4] + accQ[5] + accQ[6] + accQ[7]) * 0.0625f;

    // wave32 butterfly reduction (deterministic)
    for (int m = 16; m >= 1; m >>= 1) {
        ls += __shfl_xor(ls, m, 32);
        lq += __shfl_xor(lq, m, 32);
    }
    if (lane == 0) { s_sum[wid] = ls; s_sq[wid] = lq; }
    __syncthreads();
    if (threadIdx.x == 0) {
        float bs = 0.0f, bq = 0.0f;
        for (int i = 0; i < wavesPerBlock; ++i) { bs += s_sum[i]; bq += s_sq[i]; }
        ws[2 + 2 * blockIdx.x + 0] = bs;
        ws[2 + 2 * blockIdx.x + 1] = bq;
    }
}

// ---------------------------------------------------------------------------
// Stage 2: fold block partials (fixed order) into obsstd (ddof=1).
// ---------------------------------------------------------------------------
__global__ void stats_finalize_kernel(float* __restrict__ ws,
                                      const int* __restrict__ spinp,
                                      const int* __restrict__ trainp,
                                      int nblocks) {
    double s = 0.0, q = 0.0;
    for (int i = 0; i < nblocks; ++i) {
        s += (double)ws[2 + 2 * i + 0];
        q += (double)ws[2 + 2 * i + 1];
    }
    const double n = (double)(trainp[0] - spinp[0]);
    double var = (q - s * s / n) / (n - 1.0);
    if (!(var > 0.0)) var = 0.0;
    ws[0] = (float)sqrt(var);
}

// ---------------------------------------------------------------------------
// Parallel prologue: everything that does not depend on the recurrence state.
// Output layout (floats): 0:h_n 1B:c_n 2B:l_n 3B:lc_n 4B:z 5B:z 6B:g_oo
//                         7B:g_ol 8B:g_olc 9B:g_f 10B:h_nout(B,2) 12B:obs_std
// ---------------------------------------------------------------------------
__global__ void prep_kernel(const float* __restrict__ x,
                            const float* __restrict__ w_yom,
                            const float* __restrict__ w_ylm,
                            const float* __restrict__ w_yfm,
                            const float* __restrict__ b0,
                            const float* __restrict__ wb2,
                            const float* __restrict__ ws,
                            float* out, int B) {
    const int t = blockIdx.x * blockDim.x + threadIdx.x;
    if (t >= B) return;
    const float eo = expf(w_yom[0]);
    const float el = expf(w_ylm[0]);
    const float ef = expf(w_yfm[0]);
    const float den = eo + el + ef;
    const float oo  = eo / den;
    const float ol1 = el / den;
    const float obsstd = ws[0];

    const float u2  = x[2 * t + 1];
    const float ol3 = b0[0] + ((u2 - C_ML) / C_SL) * wb2[0];
    const float ol  = ol1 / (1.0f + expf(-ol3));

    out[(size_t)4 * B + t]  = 0.0f;            // zeros
    out[(size_t)5 * B + t]  = 0.0f;            // zeros
    out[(size_t)6 * B + t]  = oo;              // g_oo
    out[(size_t)7 * B + t]  = ol;              // g_ol (also consumed by scan)
    out[(size_t)10 * B + 2 * t + 1] = obsstd;  // h_nout[:,1]
    out[(size_t)12 * B + t] = obsstd;          // obs_std
}

// ---------------------------------------------------------------------------
// Serial scan: the irreducible nonlinear recurrence (latency-bound).
// Critical chain per step: rcp-mul (fast div) -> sub -> v_exp (TRANS) ->
// sub/select -> relu -> fma. Unrolled x4 with up-front register preloads so
// all loads sit off the dependency chain. Writes only c_n, g_olc, g_f.
// ---------------------------------------------------------------------------
__device__ __forceinline__ float scan_step(float& c0, float u1, float u2,
                                           float ol, float oo,
                                           float& olc_r_out, float& f_out) {
    const bool  pos   = c0 > 0.0f;
    const float csafe = pos ? c0 : 1.0f;
    const float d     = ol - __fdividef(u2, csafe);     // rcp + mul
    const float e     = (d > 0.0f) ? d : (__expf(d) - 1.0f);  // branchless elu
    const float olc   = pos ? (ol - e) : ol;
    float f = 1.0f - oo - olc;
    f = (f > 0.0f) ? f : 0.0f;
    const float c_prev = c0;
    c0 = fmaf(f, c0, u1);
    olc_r_out = (olc > 0.0f) ? olc : 0.0f;
    f_out = f;
    return c_prev;
}

__global__ void scan_kernel(const float* __restrict__ x,
                            const float* __restrict__ w_yom,
                            const float* __restrict__ w_ylm,
                            const float* __restrict__ w_yfm,
                            float* out, int B) {
    const float eo = expf(w_yom[0]);
    const float el = expf(w_ylm[0]);
    const float ef = expf(w_yfm[0]);
    const float oo = eo / (eo + el + ef);

    const float* ol_arr = out + (size_t)7 * B;
    float* c_n   = out + (size_t)1 * B;
    float* g_olc = out + (size_t)8 * B;
    float* g_f   = out + (size_t)9 * B;

    float c0 = 0.0f;
    int t = 0;
    for (; t + 4 <= B; t += 4) {
        // batch all loads up front (off the dependency chain)
        const float u1a = x[2 * (t + 0) + 0], u2a = x[2 * (t + 0) + 1];
        const float u1b = x[2 * (t + 1) + 0], u2b = x[2 * (t + 1) + 1];
        const float u1c = x[2 * (t + 2) + 0], u2c = x[2 * (t + 2) + 1];
        const float u1d = x[2 * (t + 3) + 0], u2d = x[2 * (t + 3) + 1];
        const float ola = ol_arr[t + 0];
        const float olb = ol_arr[t + 1];
        const float olc_ = ol_arr[t + 2];
        const float old_ = ol_arr[t + 3];

        float ra, fa, rb, fb, rc, fc, rd, fd;
        const float ca = scan_step(c0, u1a, u2a, ola,  oo, ra, fa);
        const float cb = scan_step(c0, u1b, u2b, olb,  oo, rb, fb);
        const float cc = scan_step(c0, u1c, u2c, olc_, oo, rc, fc);
        const float cd = scan_step(c0, u1d, u2d, old_, oo, rd, fd);

        c_n[t + 0] = ca; c_n[t + 1] = cb; c_n[t + 2] = cc; c_n[t + 3] = cd;
        g_olc[t + 0] = ra; g_olc[t + 1] = rb; g_olc[t + 2] = rc; g_olc[t + 3] = rd;
        g_f[t + 0] = fa; g_f[t + 1] = fb; g_f[t + 2] = fc; g_f[t + 3] = fd;
    }
    for (; t < B; ++t) {
        float r, f;
        const float c_prev = scan_step(c0, x[2 * t + 0], x[2 * t + 1],
                                       ol_arr[t], oo, r, f);
        c_n[t] = c_prev; g_olc[t] = r; g_f[t] = f;
    }
}

// ---------------------------------------------------------------------------
// Parallel epilogue: c-dependent products.
// ---------------------------------------------------------------------------
__global__ void finish_kernel(const float* __restrict__ w_yom,
                              const float* __restrict__ w_ylm,
                              const float* __restrict__ w_yfm,
                              float* out, int B) {
    const int t = blockIdx.x * blockDim.x + threadIdx.x;
    if (t >= B) return;
    const float eo = expf(w_yom[0]);
    const float el = expf(w_ylm[0]);
    const float ef = expf(w_yfm[0]);
    const float oo = eo / (eo + el + ef);

    const float c = out[(size_t)1 * B + t];
    const float h = oo * c;
    out[t]                      = h;                          // h_n
    out[(size_t)2 * B + t]      = out[(size_t)7 * B + t] * c; // l_n  = ol * c
    out[(size_t)3 * B + t]      = out[(size_t)8 * B + t] * c; // lc_n = olc_r * c
    out[(size_t)10 * B + 2 * t] = h;                          // h_nout[:,0]
}

// ---------------------------------------------------------------------------
extern "C" void kernel_launch(void* const* d_in, const int* in_sizes, int n_in,
                              void* d_out, int out_size, void* d_ws, size_t ws_size,
                              hipStream_t stream) {
    const float* x     = (const float*)d_in[0];
    const float* y     = (const float*)d_in[1];
    const float* w_yom = (const float*)d_in[2];
    const float* w_ylm = (const float*)d_in[3];
    const float* w_yfm = (const float*)d_in[4];
    const float* b0    = (const float*)d_in[5];
    const float* wb2   = (const float*)d_in[6];
    const int*   spin  = (const int*)d_in[9];
    const int*   train = (const int*)d_in[10];

    const int B = in_sizes[1];           // y_obs has B elements
    float* out = (float*)d_out;
    float* ws  = (float*)d_ws;

    const int STATS_BLOCKS = 32;         // 256 waves total, grid-stride
    stats_partial_kernel<<<STATS_BLOCKS, 256, 0, stream>>>(y, spin, train, ws);
    stats_finalize_kernel<<<1, 1, 0, stream>>>(ws, spin, train, STATS_BLOCKS);

    const int nb = (B + 255) / 256;
    prep_kernel<<<nb, 256, 0, stream>>>(x, w_yom, w_ylm, w_yfm, b0, wb2, ws, out, B);
    scan_kernel<<<1, 1, 0, stream>>>(x, w_yom, w_ylm, w_yfm, out, B);
    finish_kernel<<<nb, 256, 0, stream>>>(w_yom, w_ylm, w_yfm, out, B);
}